// FlowOps_57131654971727
// MI455X (gfx1250) — compile-verified
//
#include <hip/hip_runtime.h>

#define DN 192
#define HN 192
#define WN 192
#define HW (HN * WN)
#define NVOX (DN * HN * WN)

#define TX 64
#define TY 8
#define ZCHUNK 24
#define PR (TY + 6)          // 14 padded rows
#define PC (TX + 6)          // 70 padded cols
#define PLANE (PR * PC)      // 980 per channel
#define PLANE3 (3 * PLANE)   // 2940
#define NTHREADS (TX * TY)   // 512 = 16 waves
#define NISSUE 6             // ceil(PLANE3 / NTHREADS)

// Low 32 bits of a generic pointer into LDS are the DS-style offset (ISA 10.2).
__device__ __forceinline__ unsigned lds_off(const void* p) {
    return (unsigned)(unsigned long long)p;
}

// Issue one plane's async global->LDS copies from precomputed offsets.
// GVS addressing: 64-bit SGPR base + per-lane 32-bit byte offset.
__device__ __forceinline__ void issue_plane(unsigned lbase,
                                            const float* __restrict__ in,
                                            const unsigned* boffk,
                                            unsigned zoff, bool tail)
{
#pragma unroll
    for (int k = 0; k < NISSUE; ++k) {
        if (k < NISSUE - 1 || tail) {
            unsigned l = lbase + (unsigned)(k * NTHREADS * 4);
            unsigned b = boffk[k] + zoff;
            asm volatile("global_load_async_to_lds_b32 %0, %1, %2"
                         :: "v"(l), "v"(b), "s"(in) : "memory");
        }
    }
}

__global__ void __launch_bounds__(NTHREADS)
flowops_kernel(const float* __restrict__ in,
               const float* __restrict__ dxp,
               const float* __restrict__ dyp,
               const float* __restrict__ dzp,
               float* __restrict__ out)
{
    __shared__ float tile[2][3][PR][PC];   // double-buffered padded plane

    const int tx = threadIdx.x;            // 0..63  (W)
    const int ty = threadIdx.y;            // 0..7   (H)
    const int tid = ty * TX + tx;
    const int x0 = blockIdx.x * TX;
    const int y0 = blockIdx.y * TY;
    const int z0 = blockIdx.z * ZCHUNK;
    const int x = x0 + tx;
    const int y = y0 + ty;
    const int colbase = y * WN + x;

    const float inv_dx = 1.0f / dxp[0];    // derivative along D (axis 2)
    const float inv_dy = 1.0f / dyp[0];    // derivative along H (axis 3)
    const float inv_dz = 1.0f / dzp[0];    // derivative along W (axis 4)

    // ---- Precompute z-invariant async-copy source offsets (hoisted out of z loop).
    unsigned boffk[NISSUE];
#pragma unroll
    for (int k = 0; k < NISSUE; ++k) {
        int i = tid + k * NTHREADS;
        if (i < PLANE3) {
            int ch  = i / PLANE;
            int rem = i - ch * PLANE;
            int r   = rem / PC;
            int c   = rem - r * PC;
            int gy  = y0 - 3 + r; gy = gy < 0 ? 0 : (gy > HN - 1 ? HN - 1 : gy);
            int gx  = x0 - 3 + c; gx = gx < 0 ? 0 : (gx > WN - 1 ? WN - 1 : gx);
            boffk[k] = (unsigned)((ch * NVOX + gy * WN + gx) * 4);
        } else {
            boffk[k] = 0u;
        }
    }
    const bool tail = tid < (PLANE3 - (NISSUE - 1) * NTHREADS);   // tid < 380
    const unsigned lb0 = lds_off(&tile[0][0][0][0]) + 4u * (unsigned)tid;
    const unsigned lb1 = lds_off(&tile[1][0][0][0]) + 4u * (unsigned)tid;

    // Prologue: async-stage plane z0 into buffer 0.
    issue_plane(lb0, in, boffk, (unsigned)(z0 * HW * 4), tail);

    // Prefill the z (D-axis) register window: wnd[c][k] = in[c][clamp(z0+k-3)][y][x]
    float wnd[3][7];
#pragma unroll
    for (int c = 0; c < 3; ++c) {
#pragma unroll
        for (int k = 0; k < 7; ++k) {
            int zz = z0 + k - 3; zz = zz < 0 ? 0 : (zz > DN - 1 ? DN - 1 : zz);
            wnd[c][k] = in[c * NVOX + zz * HW + colbase];
        }
    }

    asm volatile("s_wait_asynccnt 0" ::: "memory");
    __syncthreads();

    int cur = 0;
    for (int zi = 0; zi < ZCHUNK; ++zi) {
        const int z = z0 + zi;

        // Kick off async staging of the NEXT plane while we compute this one.
        if (zi + 1 < ZCHUNK)
            issue_plane(cur ? lb0 : lb1, in, boffk,
                        (unsigned)((z + 1) * HW * 4), tail);

        // Prefetch next window slab early (hides latency behind LDS compute).
        int zn = z + 4 > DN - 1 ? DN - 1 : z + 4;
        float nw0 = in[0 * NVOX + zn * HW + colbase];
        float nw1 = in[1 * NVOX + zn * HW + colbase];
        float nw2 = in[2 * NVOX + zn * HW + colbase];

        const float c1 = 0.75f, c2 = -3.0f / 20.0f, c3 = 1.0f / 60.0f;
        float dD[3], dH[3], dWd[3];
#pragma unroll
        for (int c = 0; c < 3; ++c) {
            const float* rc = &tile[cur][c][ty + 3][tx + 3];
            // W-derivative (contiguous cols, conflict-free LDS)
            float a1 = rc[1],      b1 = rc[-1];
            float a2 = rc[2],      b2 = rc[-2];
            float a3 = rc[3],      b3 = rc[-3];
            dWd[c] = (c1 * (a1 - b1) + c2 * (a2 - b2) + c3 * (a3 - b3)) * inv_dz;
            // H-derivative (row stride PC=70, lanes hit consecutive banks)
            float r1 = rc[PC],     s1 = rc[-PC];
            float r2 = rc[2 * PC], s2 = rc[-2 * PC];
            float r3 = rc[3 * PC], s3 = rc[-3 * PC];
            dH[c] = (c1 * (r1 - s1) + c2 * (r2 - s2) + c3 * (r3 - s3)) * inv_dy;
            // D-derivative from the register sliding window
            dD[c] = (c1 * (wnd[c][4] - wnd[c][2]) +
                     c2 * (wnd[c][5] - wnd[c][1]) +
                     c3 * (wnd[c][6] - wnd[c][0])) * inv_dx;
        }

        float u_x = dD[0],  v_x = dD[1],  w_x = dD[2];
        float u_y = dH[0],  v_y = dH[1],  w_y = dH[2];
        float u_z = dWd[0], v_z = dWd[1], w_z = dWd[2];

        float sxy = 0.5f * (u_y + v_x);
        float sxz = 0.5f * (u_z + w_x);
        float syz = 0.5f * (w_y + v_z);
        float strain = u_x * u_x + v_y * v_y + w_z * w_z +
                       2.0f * (sxy * sxy + sxz * sxz + syz * syz);
        float ox = w_y - v_z;
        float oy = u_z - w_x;
        float oz = v_x - u_y;
        float ens  = ox * ox + oy * oy + oz * oz;
        float cont = u_x + v_y + w_z;

        const long off = (long)z * HW + colbase;
        __builtin_nontemporal_store(strain, out + off);
        __builtin_nontemporal_store(ox,   out + (long)NVOX + off);
        __builtin_nontemporal_store(oy,   out + 2L * NVOX + off);
        __builtin_nontemporal_store(oz,   out + 3L * NVOX + off);
        __builtin_nontemporal_store(ens,  out + 4L * NVOX + off);
        __builtin_nontemporal_store(cont, out + 5L * NVOX + off);

        // Slide the z window.
#pragma unroll
        for (int c = 0; c < 3; ++c) {
#pragma unroll
            for (int k = 0; k < 6; ++k) wnd[c][k] = wnd[c][k + 1];
        }
        wnd[0][6] = nw0; wnd[1][6] = nw1; wnd[2][6] = nw2;

        // Own async loads done, then make them visible to all waves.
        asm volatile("s_wait_asynccnt 0" ::: "memory");
        __syncthreads();
        cur ^= 1;
    }
}

extern "C" void kernel_launch(void* const* d_in, const int* in_sizes, int n_in,
                              void* d_out, int out_size, void* d_ws, size_t ws_size,
                              hipStream_t stream) {
    const float* in  = (const float*)d_in[0];
    const float* dx  = (const float*)d_in[1];
    const float* dy  = (const float*)d_in[2];
    const float* dz  = (const float*)d_in[3];
    float* out = (float*)d_out;
    (void)in_sizes; (void)n_in; (void)out_size; (void)d_ws; (void)ws_size;

    dim3 grid(WN / TX, HN / TY, DN / ZCHUNK);   // 3 x 24 x 8 = 576 blocks
    dim3 block(TX, TY, 1);                      // 512 threads = 16 wave32s
    hipLaunchKernelGGL(flowops_kernel, grid, block, 0, stream, in, dx, dy, dz, out);
}